// SwitchRouterLoss_13125420057240
// MI455X (gfx1250) — compile-verified
//
#include <hip/hip_runtime.h>

#define GROUPS 64
#define TOKENS 8192
#define NEXP   64
#define CHUNKS 8
#define TOK_PER_BLOCK (TOKENS / CHUNKS)       // 1024
#define THREADS 128
#define WAVES (THREADS / 32)                  // 4
#define TOK_PER_LANE (TOK_PER_BLOCK / THREADS) // 8
#define NBLOCKS (GROUPS * CHUNKS)             // 512
#define STAGE_PAD 34                          // even pad -> b64-mergeable, conflict-free

typedef __attribute__((ext_vector_type(2))) float v2f;
typedef __attribute__((ext_vector_type(8))) float v8f;

__global__ __launch_bounds__(THREADS)
void router_pass1(const float* __restrict__ logits,
                  float* __restrict__ psum_part,   // [NBLOCKS][NEXP]
                  int*   __restrict__ hist_part,   // [NBLOCKS][NEXP]
                  float* __restrict__ zsq_part)    // [NBLOCKS]
{
    const int blk  = blockIdx.x;
    const int g    = blk / CHUNKS;
    const int c    = blk % CHUNKS;
    const int tid  = threadIdx.x;
    const int wave = tid >> 5;
    const int lane = tid & 31;

    __shared__ int   s_hist[NEXP];
    __shared__ float s_stage[WAVES][NEXP][STAGE_PAD]; // [e][lane] column-major: conflict-free
    __shared__ float s_part[WAVES][NEXP];
    __shared__ float s_z[WAVES];

    if (tid < NEXP) s_hist[tid] = 0;
    __syncthreads();

    const float* base = logits + ((size_t)g * TOKENS + (size_t)c * TOK_PER_BLOCK) * NEXP;

    float acc[NEXP];
#pragma unroll
    for (int e = 0; e < NEXP; ++e) acc[e] = 0.0f;
    float zacc = 0.0f;

    for (int it = 0; it < TOK_PER_LANE; ++it) {
        const int tok = it * THREADS + tid;                 // block-contiguous tokens
        const float4* p4 = (const float4*)(base + (size_t)tok * NEXP);
        float l[NEXP];
#pragma unroll
        for (int i = 0; i < NEXP / 4; ++i) {                // 16x global_load_b128, deep MLP
            float4 v = p4[i];
            l[4*i+0] = v.x; l[4*i+1] = v.y; l[4*i+2] = v.z; l[4*i+3] = v.w;
        }
        float m = -3.402823466e38f;
        int   am = 0;
#pragma unroll
        for (int e = 0; e < NEXP; ++e) {
            if (l[e] > m) { m = l[e]; am = e; }             // first-max == jnp.argmax ties
        }
        float s = 0.0f;
#pragma unroll
        for (int e = 0; e < NEXP; ++e) {
            float ex = __expf(l[e] - m);
            l[e] = ex;
            s += ex;
        }
        const float inv = 1.0f / s;
#pragma unroll
        for (int e = 0; e < NEXP; ++e) acc[e] = fmaf(l[e], inv, acc[e]);
        const float lz = m + __logf(s);                     // logsumexp
        zacc = fmaf(lz, lz, zacc);
        atomicAdd(&s_hist[am], 1);                          // LDS int atomic (deterministic)
    }

    // ---- stage per-lane prob accumulators for the WMMA cross-lane reduction ----
#pragma unroll
    for (int e = 0; e < NEXP; ++e) s_stage[wave][e][lane] = acc[e];

    // wave-level z reduction (butterfly)
#pragma unroll
    for (int off = 16; off > 0; off >>= 1) zacc += __shfl_xor(zacc, off, 32);
    if (lane == 0) s_z[wave] = zacc;

    __syncthreads();

    // ---- per-wave column sum over 32 lanes via V_WMMA_F32_16X16X4_F32 ----
    // D = ones(16x4) x B(4x16), accumulated over 8 K-chunks (32 lanes total).
    // B operand layout: lanes 0-15 hold {K0,K1} for N=lane; lanes 16-31 hold {K2,K3}.
    const v2f a_ones = {1.0f, 1.0f};
    const int n  = lane & 15;
    const int hi = lane >> 4;
#pragma unroll
    for (int eg = 0; eg < 4; ++eg) {                        // 4 groups of 16 experts
        v8f cacc = {};
#pragma unroll
        for (int k = 0; k < 8; ++k) {                       // K-chunk = lanes 4k..4k+3
            v2f b;
            b[0] = s_stage[wave][eg * 16 + n][4 * k + 2 * hi + 0];
            b[1] = s_stage[wave][eg * 16 + n][4 * k + 2 * hi + 1];
            cacc = __builtin_amdgcn_wmma_f32_16x16x4_f32(
                false, a_ones, false, b, (short)0, cacc, false, false);
        }
        // C layout: VGPR0, lanes 0-15 = row M=0, N=lane -> the 16 expert sums
        if (hi == 0) s_part[wave][eg * 16 + n] = cacc[0];
    }
    __syncthreads();

    // fixed-order cross-wave combine -> deterministic partials to workspace
    if (tid < NEXP) {
        float s = 0.0f;
#pragma unroll
        for (int w = 0; w < WAVES; ++w) s += s_part[w][tid];
        psum_part[(size_t)blk * NEXP + tid] = s;
        hist_part[(size_t)blk * NEXP + tid] = s_hist[tid];
    }
    if (tid == 0) {
        float z = 0.0f;
        for (int w = 0; w < WAVES; ++w) z += s_z[w];
        zsq_part[blk] = z;
    }
}

__global__ __launch_bounds__(NEXP)
void router_pass2(const float* __restrict__ psum_part,
                  const int*   __restrict__ hist_part,
                  const float* __restrict__ zsq_part,
                  const int*   __restrict__ cap_ptr,
                  float* __restrict__ out)
{
    const int e   = threadIdx.x;   // 64 threads, one per expert
    const int cap = *cap_ptr;
    __shared__ int    s_excess[NEXP];
    __shared__ int    s_drop;
    __shared__ double s_red[NEXP];

    double auxacc = 0.0;
    for (int g = 0; g < GROUPS; ++g) {
        int   cnt = 0;
        float ps  = 0.0f;
        for (int c = 0; c < CHUNKS; ++c) {
            const size_t idx = ((size_t)(g * CHUNKS + c)) * NEXP + e;
            cnt += hist_part[idx];
            ps  += psum_part[idx];
        }
        const int kept = cnt < cap ? cnt : cap;
        s_excess[e] = cnt - kept;
        __syncthreads();
        if (e == 0) {
            int d = 0;
            for (int i = 0; i < NEXP; ++i) d += s_excess[i];
            s_drop = d;                 // dropped tokens all argmax to expert 0
        }
        __syncthreads();
        const int fin = kept + (e == 0 ? s_drop : 0);
        auxacc += (double)fin * (double)ps;
        __syncthreads();
    }
    s_red[e] = auxacc;
    __syncthreads();
    if (e == 0) {
        double aux_sum = 0.0;
        for (int i = 0; i < NEXP; ++i) aux_sum += s_red[i];
        double z_sum = 0.0;
        for (int b = 0; b < NBLOCKS; ++b) z_sum += (double)zsq_part[b];
        const double z_loss = z_sum / ((double)GROUPS * (double)TOKENS);
        // aux = mean_{g,e}(cnt/T * ps/T) * E^2 = aux_sum * E / (G*T*T)
        const double aux = aux_sum * (double)NEXP /
                           ((double)GROUPS * (double)TOKENS * (double)TOKENS);
        out[0] = (float)(1.0e-3 * z_loss + 1.0e-3 * aux);
    }
}

extern "C" void kernel_launch(void* const* d_in, const int* in_sizes, int n_in,
                              void* d_out, int out_size, void* d_ws, size_t ws_size,
                              hipStream_t stream)
{
    const float* logits = (const float*)d_in[0];
    // d_in[1] = attention_mask (unused by the forward computation)
    const int* cap = (const int*)d_in[2];
    float* out = (float*)d_out;

    char* ws = (char*)d_ws;
    float* psum = (float*)ws;                                           // 512*64*4 = 128 KiB
    int*   hist = (int*)(ws + (size_t)NBLOCKS * NEXP * sizeof(float));  // 128 KiB
    float* zsq  = (float*)(ws + 2 * (size_t)NBLOCKS * NEXP * sizeof(float)); // 2 KiB

    hipLaunchKernelGGL(router_pass1, dim3(NBLOCKS), dim3(THREADS), 0, stream,
                       logits, psum, hist, zsq);
    hipLaunchKernelGGL(router_pass2, dim3(1), dim3(NEXP), 0, stream,
                       psum, hist, zsq, cap, out);
}